// YoloLoss_88983132438568
// MI455X (gfx1250) — compile-verified
//
#include <hip/hip_runtime.h>
#include <hip/hip_bf16.h>
#include <math.h>

// ---- problem constants (fixed by the reference) ----
#define HH 64
#define WW 64
#define HWD 4096          // H*W
#define AA 5              // anchors
#define CC 20             // classes
#define GG 32             // GT boxes per image
#define CHN 25            // 5 + C
#define INV_RED (1.0f/32.0f)
#define IOU_THR 0.6f
#define OBJ_SCALE 5.0f

typedef __attribute__((ext_vector_type(2))) float v2f;
typedef __attribute__((ext_vector_type(8))) float v8f;

__device__ __forceinline__ float sigm(float x) { return 1.0f / (1.0f + __expf(-x)); }

// ds_swizzle_b32 group-of-32 XOR swap: offset = {xor[4:0]<<10, or=0, and=0x1F}
template <int OFF>
__device__ __forceinline__ float swz(float v) {
  return __int_as_float(__builtin_amdgcn_ds_swizzle(__float_as_int(v), OFF));
}

// Wave-level fp32 sum using V_WMMA_F32_16X16X4_F32.
// A = ones(16x4), B holds each lane's partial exactly once (b[0]=v, b[1]=0).
// D[m][n] = colsum(n) for every m; lanes 0..15 hold colsum(n=lane) in c[0]
// (duplicated in lanes 16..31), so XOR-swizzle sums over the 16 distinct
// columns finish the reduction. fp32-exact accumulation throughout.
__device__ __forceinline__ float wave_sum_wmma(float v) {
  v2f a; a[0] = 1.0f; a[1] = 1.0f;
  v2f b; b[0] = v;    b[1] = 0.0f;
  v8f c = {};
  c = __builtin_amdgcn_wmma_f32_16x16x4_f32(false, a, false, b, (short)0, c, false, false);
  float col = c[0];
  col += swz<0x041F>(col);   // xor 1
  col += swz<0x081F>(col);   // xor 2
  col += swz<0x101F>(col);   // xor 4
  col += swz<0x201F>(col);   // xor 8
  return col;   // every lane holds the full 32-lane sum
}

// ---- stage 1: per-GT setup (B*G = 512 items) ----
__global__ void yolo_setup(const float* __restrict__ target,
                           const unsigned char* __restrict__ gt_valid,
                           const float* __restrict__ anchors,
                           float* __restrict__ ws_gt,  // [BG][4] gx,gy,gw,gh
                           float* __restrict__ ws_tc,  // [BG][4] tx,ty,tw,th
                           int*   __restrict__ ws_flat,// [BG] flat idx or -1
                           int*   __restrict__ ws_tcls,// [BG]
                           int BG) {
  int i = blockIdx.x * blockDim.x + threadIdx.x;
  if (i >= BG) return;
  int b = i / GG;
  const float* t = target + (size_t)i * 5;
  float gx = (t[0] + 0.5f * t[2]) * INV_RED;
  float gy = (t[1] + 0.5f * t[3]) * INV_RED;
  float gw = t[2] * INV_RED;
  float gh = t[3] * INV_RED;

  // argmax IoU vs centered anchors (wh only): inter = min(w)*min(h)
  int   best = 0;
  float bi   = -1.0f;
#pragma unroll
  for (int n = 0; n < AA; ++n) {
    float aw = anchors[2 * n], ah = anchors[2 * n + 1];
    float inter = fminf(gw, aw) * fminf(gh, ah);
    float iou   = inter / (gw * gh + aw * ah - inter);
    if (iou > bi) { bi = iou; best = n; }
  }
  int gi = min(max((int)gx, 0), WW - 1);
  int gj = min(max((int)gy, 0), HH - 1);
  int cell = gj * WW + gi;
  float aw = anchors[2 * best], ah = anchors[2 * best + 1];

  ws_gt[4 * i + 0] = gx;  ws_gt[4 * i + 1] = gy;
  ws_gt[4 * i + 2] = gw;  ws_gt[4 * i + 3] = gh;
  ws_tc[4 * i + 0] = gx - (float)gi;
  ws_tc[4 * i + 1] = gy - (float)gj;
  ws_tc[4 * i + 2] = __logf(fmaxf(gw, 1.0f) / aw);
  ws_tc[4 * i + 3] = __logf(fmaxf(gh, 1.0f) / ah);
  ws_tcls[i] = (int)t[4];
  ws_flat[i] = (gt_valid[i] != 0) ? (b * (AA * HWD) + best * HWD + cell) : -1;
}

// ---- stage 2: fused per-cell loss (B*A*HW threads) ----
__global__ __launch_bounds__(256, 4)
void yolo_main(const float* __restrict__ out,
               const float* __restrict__ anchors,
               const float* __restrict__ ws_gt,
               const float* __restrict__ ws_tc,
               const int*   __restrict__ ws_flat,
               const int*   __restrict__ ws_tcls,
               float* __restrict__ ws_part) {
  __shared__ float s_gt[GG * 4];
  __shared__ float s_tc[GG * 4];
  __shared__ int   s_flat[GG];
  __shared__ int   s_tcls[GG];
  __shared__ float s_red[8];

  const int b   = blockIdx.z;
  const int a   = blockIdx.y;
  const int tid = threadIdx.x;

  if (tid < GG) {
    int s = b * GG + tid;
    ((float4*)s_gt)[tid] = ((const float4*)ws_gt)[s];
    ((float4*)s_tc)[tid] = ((const float4*)ws_tc)[s];
    s_flat[tid] = ws_flat[s];
    s_tcls[tid] = ws_tcls[s];
  }
  __syncthreads();

  const int hw = blockIdx.x * 256 + tid;
  const float* base = out + ((size_t)(b * AA + a) * CHN) * HWD + hw;

  // decode prediction (coalesced: 5 channel-strided dword streams)
  float tx = base[0 * HWD];
  float ty = base[1 * HWD];
  float tw = base[2 * HWD];
  float th = base[3 * HWD];
  float tc = base[4 * HWD];

  float sx = sigm(tx), sy = sigm(ty);
  float aw = anchors[2 * a], ah = anchors[2 * a + 1];   // uniform -> scalar loads
  float pw = __expf(tw) * aw, ph = __expf(th) * ah;
  float px = sx + (float)(hw & (WW - 1));
  float py = sy + (float)(hw >> 6);
  float px1 = px - 0.5f * pw, px2 = px + 0.5f * pw;
  float py1 = py - 0.5f * ph, py2 = py + 0.5f * ph;
  float parea = pw * ph;

  const int myflat = (b * AA + a) * HWD + hw;
  int   match  = -1;
  float tconf  = 0.0f;
  bool  ignore = false;

#pragma unroll 8
  for (int g = 0; g < GG; ++g) {
    float gx = s_gt[4 * g + 0], gy = s_gt[4 * g + 1];
    float gw = s_gt[4 * g + 2], gh = s_gt[4 * g + 3];
    float gx1 = gx - 0.5f * gw, gx2 = gx + 0.5f * gw;
    float gy1 = gy - 0.5f * gh, gy2 = gy + 0.5f * gh;
    float dx = fmaxf(fminf(px2, gx2) - fmaxf(px1, gx1), 0.0f);
    float dy = fmaxf(fminf(py2, gy2) - fmaxf(py1, gy1), 0.0f);
    float inter = dx * dy;
    float iou = inter / (parea + gw * gh - inter);
    int fl = s_flat[g];                      // -1 when GT invalid
    if (fl >= 0 && iou > IOU_THR) ignore = true;
    if (fl == myflat) { match = g; tconf = iou; }   // last writer wins (scatter-set)
  }

  float conf = sigm(tc);
  float part;
  if (match >= 0) {
    float d = conf - tconf;
    part = OBJ_SCALE * d * d;                        // conf_mask = OBJECT_SCALE
    // coord loss at this cell
    float c0 = sx - s_tc[4 * match + 0];
    float c1 = sy - s_tc[4 * match + 1];
    float c2 = tw - s_tc[4 * match + 2];
    float c3 = th - s_tc[4 * match + 3];
    part += c0 * c0 + c1 * c1 + c2 * c2 + c3 * c3;
    // class CE at this cell (rare: gather 20 strided logits, L0-resident)
    const float* cl = base + 5 * HWD;
    float m = -1e30f;
#pragma unroll
    for (int c = 0; c < CC; ++c) m = fmaxf(m, cl[(size_t)c * HWD]);
    float se = 0.0f;
#pragma unroll
    for (int c = 0; c < CC; ++c) se += __expf(cl[(size_t)c * HWD] - m);
    float lse = m + __logf(se);
    part += 2.0f * (lse - cl[(size_t)s_tcls[match] * HWD]);
  } else {
    part = ignore ? 0.0f : conf * conf;              // tconf=0, mask 0 or 1
  }

  // wave32 sum via fp32 WMMA, then 8-wave LDS combine -> one partial per block
  float wsum = wave_sum_wmma(part);
  if ((tid & 31) == 0) s_red[tid >> 5] = wsum;
  __syncthreads();
  if (tid == 0) {
    float s = 0.0f;
#pragma unroll
    for (int i = 0; i < 8; ++i) s += s_red[i];
    ws_part[(blockIdx.z * gridDim.y + blockIdx.y) * gridDim.x + blockIdx.x] = s;
  }
}

// ---- stage 3: reduce block partials to the scalar loss ----
__global__ void yolo_finish(const float* __restrict__ ws_part, int nparts,
                            float invB, float* __restrict__ out) {
  __shared__ float s_red[8];
  int tid = threadIdx.x;
  float s = 0.0f;
  for (int i = tid; i < nparts; i += 256) s += ws_part[i];
  float wsum = wave_sum_wmma(s);
  if ((tid & 31) == 0) s_red[tid >> 5] = wsum;
  __syncthreads();
  if (tid == 0) {
    float t = 0.0f;
#pragma unroll
    for (int i = 0; i < 8; ++i) t += s_red[i];
    out[0] = t * invB;
  }
}

extern "C" void kernel_launch(void* const* d_in, const int* in_sizes, int n_in,
                              void* d_out, int out_size, void* d_ws, size_t ws_size,
                              hipStream_t stream) {
  const float*         output   = (const float*)d_in[0];        // (B,125,64,64) f32
  const float*         target   = (const float*)d_in[1];        // (B,32,5) f32
  const unsigned char* gt_valid = (const unsigned char*)d_in[2];// (B,32) bool
  const float*         anchors  = (const float*)d_in[3];        // (5,2) f32

  const int BG = in_sizes[2];        // B * 32
  const int B  = BG / GG;

  // workspace layout (floats): gt[4*BG] | tc[4*BG] | flat[BG] | tcls[BG] | partials
  float* ws      = (float*)d_ws;
  float* ws_gt   = ws;
  float* ws_tc   = ws + (size_t)4 * BG;
  int*   ws_flat = (int*)(ws + (size_t)8 * BG);
  int*   ws_tcls = (int*)(ws + (size_t)9 * BG);
  float* ws_part = ws + (size_t)10 * BG;

  yolo_setup<<<(BG + 255) / 256, 256, 0, stream>>>(
      target, gt_valid, anchors, ws_gt, ws_tc, ws_flat, ws_tcls, BG);

  dim3 grid(HWD / 256, AA, B);
  yolo_main<<<grid, 256, 0, stream>>>(
      output, anchors, ws_gt, ws_tc, ws_flat, ws_tcls, ws_part);

  const int nparts = B * AA * (HWD / 256);
  yolo_finish<<<1, 256, 0, stream>>>(ws_part, nparts, 1.0f / (float)B, (float*)d_out);
}